// PointNet_51960514347353
// MI455X (gfx1250) — compile-verified
//
#include <hip/hip_runtime.h>

// Geometry constants (match reference)
#define Bd   2
#define Cc   128
#define Hd   64
#define Wd   512
#define HW   32768          // Hd*Wd
#define EPSV 1e-5f

typedef float v2f __attribute__((ext_vector_type(2)));
typedef float v8f __attribute__((ext_vector_type(8)));

// ---------------------------------------------------------------------------
// Kernel 0: fold BN running stats into scale/shift  (prm layout, floats):
//   [0..127]   scale_f     [128..255] shift_f
//   [256..383] scale_p     [384..511] shift_p
//   [512..514] scale_e     [516..518] shift_e
// ---------------------------------------------------------------------------
__global__ void prep_params(const float* __restrict__ pg, const float* __restrict__ pb,
                            const float* __restrict__ pm, const float* __restrict__ pv,
                            const float* __restrict__ fg, const float* __restrict__ fb,
                            const float* __restrict__ fm, const float* __restrict__ fv,
                            const float* __restrict__ eg, const float* __restrict__ eb,
                            const float* __restrict__ em, const float* __restrict__ ev,
                            float* __restrict__ prm) {
    int t = threadIdx.x;
    if (t < 128) {
        float sf = fg[t] / sqrtf(fv[t] + EPSV);
        prm[t]        = sf;
        prm[128 + t]  = fb[t] - fm[t] * sf;
        float sp = pg[t] / sqrtf(pv[t] + EPSV);
        prm[256 + t]  = sp;
        prm[384 + t]  = pb[t] - pm[t] * sp;
    } else if (t < 131) {
        int j = t - 128;
        float se = eg[j] / sqrtf(ev[j] + EPSV);
        prm[512 + j] = se;
        prm[516 + j] = eb[j] - em[j] * se;
    }
}

// ---------------------------------------------------------------------------
// Kernel 1: y[b,c,p] = relu(scale_f[c] * (fw @ features)[b,c,p] + shift_f[c])
// WMMA f32 16x16x4, K=128 in 32 steps. Block = 256 threads = 8 waves.
// Wave w owns channel tile [16w, 16w+16); block covers 128 consecutive
// positions (8 N-tiles of 16) so A fragments are loaded once per wave.
//
// A (16x4 f32) layout: lane<16 -> M=lane, v0=K0,v1=K1 ; lane>=16 -> M=lane-16,
//                      v0=K2,v1=K3.
// B (4x16 f32) layout (mirror): lane<16 -> N=lane, v0=row K0, v1=row K1 ;
//                      lane>=16 -> N=lane-16, v0=row K2, v1=row K3.
// C/D (16x16 f32): VGPR r, lane L -> M = r + 8*(L>=16), N = L&15.
// ---------------------------------------------------------------------------
__global__ __launch_bounds__(256) void gemm_bn_relu(
    const float* __restrict__ feat,   // (B, C, HW)
    const float* __restrict__ fw,     // (C, C)
    const float* __restrict__ prm,
    float* __restrict__ y)            // (B, C, HW)
{
    const int lane = threadIdx.x & 31;
    const int wave = threadIdx.x >> 5;
    const int c0   = wave * 16;
    const int q0   = blockIdx.x * 128;      // global position base (b*HW+p)
    const int lrow = lane & 15;
    const int kh   = (lane >> 4) * 2;       // K sub-offset for this half-wave

    // --- A fragments: fw rows c0..c0+15, full K=128 ---
    v2f a[32];
    const float* fwr = fw + (c0 + lrow) * Cc + kh;
#pragma unroll
    for (int j = 0; j < 32; ++j) {
        a[j].x = fwr[4 * j];
        a[j].y = fwr[4 * j + 1];
    }

    // --- per-lane BN params for the 8 channels this lane's C rows map to ---
    const int mBase = c0 + 8 * (lane >> 4);
    float sc[8], sh[8];
#pragma unroll
    for (int r = 0; r < 8; ++r) {
        sc[r] = prm[mBase + r];
        sh[r] = prm[128 + mBase + r];
    }

#pragma unroll 1
    for (int nt = 0; nt < 8; ++nt) {
        const int q    = q0 + nt * 16 + lrow;   // this lane's N column
        const int bidx = q >> 15;               // / HW
        const int p    = q & (HW - 1);
        const float* fcol = feat + (size_t)bidx * Cc * HW + p;

        v8f acc = {};
#pragma unroll
        for (int j = 0; j < 32; ++j) {
            v2f bf;
            const int k0 = 4 * j + kh;
            bf.x = fcol[(size_t)k0 * HW];
            bf.y = fcol[(size_t)(k0 + 1) * HW];
            acc = __builtin_amdgcn_wmma_f32_16x16x4_f32(
                /*neg_a=*/false, a[j], /*neg_b=*/false, bf,
                /*c_mod=*/(short)0, acc, /*reuse_a=*/false, /*reuse_b=*/false);
        }

        float* yb = y + (size_t)bidx * Cc * HW + p;
#pragma unroll
        for (int r = 0; r < 8; ++r) {
            float v = fmaxf(acc[r] * sc[r] + sh[r], 0.0f);
            yb[(size_t)(mBase + r) * HW] = v;
        }
    }
}

// ---------------------------------------------------------------------------
// Kernel 2: per position p:
//   geo[c] = max_n ( (y[c, p_n] + relu(sp[c]*(pw[c]·rel_n) + shp[c])) * mu_n )
//   e2c_j  = se[j] * (ew[j]·geo) + she[j];  cart_out = cart + e2c*mask
// mu_n = 0 for OOB / masked neighbors, and every term >= 0 => init max at 0.
// ---------------------------------------------------------------------------
__global__ __launch_bounds__(256) void fuse_max_out(
    const float* __restrict__ y,      // (B, C, HW)
    const float* __restrict__ cart,   // (B, 3, HW)
    const int*   __restrict__ mask,   // (B, 1, HW), 0/1
    const float* __restrict__ pw,     // (C, 3)
    const float* __restrict__ ew,     // (3, C)
    const float* __restrict__ prm,
    float* __restrict__ out)          // geo (B,C,HW) then cart_out (B,3,HW)
{
    const int q    = blockIdx.x * blockDim.x + threadIdx.x;  // 0..P-1
    const int bidx = q >> 15;
    const int p    = q & (HW - 1);
    const int h    = p >> 9;
    const int w    = p & (Wd - 1);

    const float* cb = cart + (size_t)bidx * 3 * HW;
    const float cx = cb[p], cy = cb[HW + p], cz = cb[2 * HW + p];
    const float maskc = (float)mask[bidx * HW + p];

    float rx[9], ry[9], rz[9], mu[9];
    int   yofs[9];
#pragma unroll
    for (int n = 0; n < 9; ++n) {
        const int dh = n / 3 - 1, dw = n % 3 - 1;
        const int hn = h + dh, wn = w + dw;
        const bool inb = ((unsigned)hn < (unsigned)Hd) && ((unsigned)wn < (unsigned)Wd);
        const int pn = inb ? (hn * Wd + wn) : p;   // clamped addr; zeroed by mu
        mu[n] = inb ? (float)mask[bidx * HW + pn] : 0.0f;
        rx[n] = inb ? (cb[pn] - cx) : 0.0f;
        ry[n] = inb ? (cb[HW + pn] - cy) : 0.0f;
        rz[n] = inb ? (cb[2 * HW + pn] - cz) : 0.0f;
        yofs[n] = pn;
    }

    float a0 = 0.0f, a1 = 0.0f, a2 = 0.0f;
    const float* yb = y + (size_t)bidx * Cc * HW;
    float* geo_out = out + (size_t)bidx * Cc * HW + p;

#pragma unroll 1
    for (int c = 0; c < Cc; ++c) {
        const float p0 = pw[c * 3], p1 = pw[c * 3 + 1], p2 = pw[c * 3 + 2];
        const float sp = prm[256 + c], shp = prm[384 + c];
        const float* yc = yb + (size_t)c * HW;
        float m = 0.0f;
#pragma unroll
        for (int n = 0; n < 9; ++n) {
            const float d   = p0 * rx[n] + p1 * ry[n] + p2 * rz[n];
            const float pos = fmaxf(sp * d + shp, 0.0f);
            const float t   = (yc[yofs[n]] + pos) * mu[n];
            m = fmaxf(m, t);
        }
        geo_out[(size_t)c * HW] = m;
        a0 += ew[c] * m;
        a1 += ew[Cc + c] * m;
        a2 += ew[2 * Cc + c] * m;
    }

    float* co = out + (size_t)Bd * Cc * HW + (size_t)bidx * 3 * HW;
    co[p]          = cx + (prm[512] * a0 + prm[516]) * maskc;
    co[HW + p]     = cy + (prm[513] * a1 + prm[517]) * maskc;
    co[2 * HW + p] = cz + (prm[514] * a2 + prm[518]) * maskc;
}

// ---------------------------------------------------------------------------
extern "C" void kernel_launch(void* const* d_in, const int* in_sizes, int n_in,
                              void* d_out, int out_size, void* d_ws, size_t ws_size,
                              hipStream_t stream) {
    (void)in_sizes; (void)n_in; (void)out_size; (void)ws_size;

    const float* features = (const float*)d_in[0];
    const float* cart     = (const float*)d_in[1];
    const int*   mask     = (const int*)  d_in[2];
    const float* pw       = (const float*)d_in[3];
    const float* pg       = (const float*)d_in[4];
    const float* pb       = (const float*)d_in[5];
    const float* pm       = (const float*)d_in[6];
    const float* pv       = (const float*)d_in[7];
    const float* fw       = (const float*)d_in[8];
    const float* fg       = (const float*)d_in[9];
    const float* fb       = (const float*)d_in[10];
    const float* fm       = (const float*)d_in[11];
    const float* fv       = (const float*)d_in[12];
    const float* ew       = (const float*)d_in[13];
    const float* eg       = (const float*)d_in[14];
    const float* eb       = (const float*)d_in[15];
    const float* em       = (const float*)d_in[16];
    const float* ev       = (const float*)d_in[17];

    float* ws  = (float*)d_ws;
    float* yv  = ws;                                  // (B, C, HW) = 33.5 MB
    float* prm = ws + (size_t)Bd * Cc * HW;           // 520 floats

    prep_params<<<1, 256, 0, stream>>>(pg, pb, pm, pv, fg, fb, fm, fv,
                                       eg, eb, em, ev, prm);

    // P = B*HW = 65536 positions; 128 positions per block -> 512 blocks
    gemm_bn_relu<<<512, 256, 0, stream>>>(features, fw, prm, yv);

    // one thread per position -> 256 blocks of 256
    fuse_max_out<<<256, 256, 0, stream>>>(yv, cart, mask, pw, ew, prm,
                                          (float*)d_out);
}